// KVCacheManager_48954037240384
// MI455X (gfx1250) — compile-verified
//
#include <hip/hip_runtime.h>

// Native clang vector types; 16B aligned.
typedef float f4  __attribute__((ext_vector_type(4)));
typedef int   v4i __attribute__((ext_vector_type(4)));

// Shapes fixed by the reference: L=2, B=4, H=8, M=4096, D=128, fp32.
constexpr int Lc = 2, Bc = 4, Hc = 8, Mc = 4096, Dc = 128;
constexpr unsigned D4 = Dc / 4;                                   // 32 f4 per row
constexpr unsigned CACHE_VEC4 = (unsigned)Lc * Bc * Hc * Mc * D4; // 8,388,608 f4 per cache

// Launch geometry chosen so the grid exactly tiles the stream:
// STRIDE * ITERS == CACHE_VEC4  ->  zero bounds checks, straight-line code.
constexpr unsigned THREADS = 256;
constexpr unsigned BLOCKS  = 8192;
constexpr unsigned STRIDE  = THREADS * BLOCKS;      // 2,097,152
constexpr unsigned ITERS   = CACHE_VEC4 / STRIDE;   // 4 (per cache)
static_assert(STRIDE * ITERS == CACHE_VEC4, "grid must tile the stream exactly");

// ---------------------------------------------------------------------------
// gfx1250 async global<->LDS copy path (ASYNCcnt engine). Builtin signature
// (confirmed by compiler diagnostics): (v4i AS1* gaddr, v4i AS3* lds,
// imm offset, imm cpol). 512B moved per wave per instruction, no VGPR staging.
// ---------------------------------------------------------------------------
#if defined(__has_builtin)
#if __has_builtin(__builtin_amdgcn_global_load_async_to_lds_b128) && \
    __has_builtin(__builtin_amdgcn_global_store_async_from_lds_b128)
#define KV_HAVE_ASYNC_LDS 1
#endif
#endif

#ifndef KV_HAVE_ASYNC_LDS
#define KV_HAVE_ASYNC_LDS 0
#endif

__device__ __forceinline__ void kv_wait_asynccnt0() {
#if defined(__has_builtin) && __has_builtin(__builtin_amdgcn_s_wait_asynccnt)
    __builtin_amdgcn_s_wait_asynccnt(0);
#else
    asm volatile("s_wait_asynccnt 0x0" ::: "memory");
#endif
}

#if KV_HAVE_ASYNC_LDS
// Two-step casts: generic -> AS void* (known-good) -> AS v4i*.
#define KV_GPTR(p) ((__attribute__((address_space(1))) v4i*)( \
                    (__attribute__((address_space(1))) void*)(p)))
#define KV_LPTR(p) ((__attribute__((address_space(3))) v4i*)( \
                    (__attribute__((address_space(3))) void*)(p)))

// Each thread owns 8 LDS f4 slots: 4 chunks of k + 4 chunks of v.
// LDS = 8 * 256 * 16B = 32 KB per block (of 320 KB per WGP).
__global__ __launch_bounds__(THREADS) void kv_bulk_copy_async_kernel(
    const f4* __restrict__ kc,
    const f4* __restrict__ vc,
    f4* __restrict__ out)
{
    __shared__ f4 lds[2 * ITERS][THREADS];
    const unsigned n   = CACHE_VEC4;
    const unsigned tid = threadIdx.x;
    const unsigned i0  = blockIdx.x * THREADS + tid;

    // Issue all async loads back-to-back (ASYNCcnt += 8 for this wave).
#pragma unroll
    for (unsigned u = 0; u < ITERS; ++u)
        __builtin_amdgcn_global_load_async_to_lds_b128(
            KV_GPTR(kc + i0 + u * STRIDE), KV_LPTR(&lds[u][tid]), 0, 0);
#pragma unroll
    for (unsigned u = 0; u < ITERS; ++u)
        __builtin_amdgcn_global_load_async_to_lds_b128(
            KV_GPTR(vc + i0 + u * STRIDE), KV_LPTR(&lds[ITERS + u][tid]), 0, 0);

    // Loads must land in LDS before stores read it (LDS side is unordered).
    kv_wait_asynccnt0();

    // Issue all async stores; S_ENDPGM performs an implicit wait-idle.
#pragma unroll
    for (unsigned u = 0; u < ITERS; ++u)
        __builtin_amdgcn_global_store_async_from_lds_b128(
            KV_GPTR(out + i0 + u * STRIDE), KV_LPTR(&lds[u][tid]), 0, 0);
#pragma unroll
    for (unsigned u = 0; u < ITERS; ++u)
        __builtin_amdgcn_global_store_async_from_lds_b128(
            KV_GPTR(out + n + i0 + u * STRIDE), KV_LPTR(&lds[ITERS + u][tid]), 0, 0);

    kv_wait_asynccnt0();
}
#endif // KV_HAVE_ASYNC_LDS

// ---------------------------------------------------------------------------
// Fallback path: straight-line b128 non-temporal stream.
// Exactly ITERS chunks per cache per thread, no bounds checks, no branches.
// ---------------------------------------------------------------------------
__global__ __launch_bounds__(THREADS) void kv_bulk_copy_nt_kernel(
    const f4* __restrict__ kc,
    const f4* __restrict__ vc,
    f4* __restrict__ out)
{
    const unsigned n  = CACHE_VEC4;
    const unsigned i0 = blockIdx.x * THREADS + threadIdx.x;
#pragma unroll
    for (unsigned u = 0; u < ITERS; ++u) {
        const unsigned i = i0 + u * STRIDE;
        f4 a = __builtin_nontemporal_load(kc + i);
        f4 b = __builtin_nontemporal_load(vc + i);
        __builtin_nontemporal_store(a, out + i);
        __builtin_nontemporal_store(b, out + n + i);
    }
}

// ---------------------------------------------------------------------------
// Fused scatter: overwrite row pos[b] (< seq_len) with latest[c][l,b,h,0,:].
// 2*L*B*H*(D/4) = 4096 f4 writes — negligible. Stream-ordered after bulk.
// ---------------------------------------------------------------------------
__global__ __launch_bounds__(THREADS) void kv_scatter_kernel(
    const f4* __restrict__ latest_k,
    const f4* __restrict__ latest_v,
    const int* __restrict__ position_ids, // (B,1) int32
    const int* __restrict__ seq_len_p,    // scalar int
    f4* __restrict__ out)
{
    const unsigned total = 2u * Lc * Bc * Hc * D4; // 4096
    unsigned j = blockIdx.x * blockDim.x + threadIdx.x;
    if (j >= total) return;

    const unsigned d4 = j & (D4 - 1);
    const unsigned h  = (j >> 5) & (Hc - 1);
    const unsigned b  = (j >> 8) & (Bc - 1);
    const unsigned l  = (j >> 10) & (Lc - 1);
    const unsigned c  = j >> 11;

    const int p = position_ids[b];
    const int S = seq_len_p[0];
    if (p < 0 || p >= S || p >= Mc) return;

    const f4* lat = (c == 0) ? latest_k : latest_v;
    const f4 val = lat[(((l * Bc + b) * Hc + h) << 5) + d4];

    const unsigned row = ((c * Lc + l) * Bc + b) * Hc + h;   // < 128
    const size_t   idx = ((size_t)row * Mc + (unsigned)p) * D4 + d4;
    out[idx] = val;
}

// ---------------------------------------------------------------------------
// Inputs (setup_inputs order):
//   0: k_caches (L,B,H,M,D) f32   1: v_caches (L,B,H,M,D) f32
//   2: latest_k (L,B,H,1,D) f32   3: latest_v (L,B,H,1,D) f32
//   4: position_ids (B,1) i32     5: seq_len scalar i32
// Output: (2,L,B,H,M,D) f32
// ---------------------------------------------------------------------------
extern "C" void kernel_launch(void* const* d_in, const int* in_sizes, int n_in,
                              void* d_out, int out_size, void* d_ws, size_t ws_size,
                              hipStream_t stream)
{
    (void)in_sizes; (void)n_in; (void)out_size; (void)d_ws; (void)ws_size;

    const f4*  kc  = reinterpret_cast<const f4*>(d_in[0]);
    const f4*  vc  = reinterpret_cast<const f4*>(d_in[1]);
    const f4*  lk  = reinterpret_cast<const f4*>(d_in[2]);
    const f4*  lv  = reinterpret_cast<const f4*>(d_in[3]);
    const int* pos = reinterpret_cast<const int*>(d_in[4]);
    const int* sl  = reinterpret_cast<const int*>(d_in[5]);
    f4*        out = reinterpret_cast<f4*>(d_out);

#if KV_HAVE_ASYNC_LDS
    kv_bulk_copy_async_kernel<<<BLOCKS, THREADS, 0, stream>>>(kc, vc, out);
#else
    kv_bulk_copy_nt_kernel<<<BLOCKS, THREADS, 0, stream>>>(kc, vc, out);
#endif

    const unsigned scatter_total = 2u * Lc * Bc * Hc * D4;
    kv_scatter_kernel<<<(scatter_total + THREADS - 1) / THREADS, THREADS, 0, stream>>>(
        lk, lv, pos, sl, out);
}